// AudioOnlySpecAugment_Old_40853728920184
// MI455X (gfx1250) — compile-verified
//
#include <hip/hip_runtime.h>
#include <stdint.h>

// ---------------- problem constants ----------------
constexpr int kB = 32;
constexpr int kT = 1024;
constexpr int kD = 1536;
constexpr int kA = 1280;          // AUDIO_DIM
constexpr int kV = kD - kA;       // 256 vision features (untouched)

typedef float v4f __attribute__((ext_vector_type(4)));
typedef int   v4i __attribute__((ext_vector_type(4)));

// ---------------- JAX threefry2x32 PRNG ----------------
__device__ __forceinline__ uint32_t rotl32(uint32_t x, int r) {
  return (x << r) | (x >> (32 - r));
}
__device__ __forceinline__ void tf_r(uint32_t& x0, uint32_t& x1, int r) {
  x0 += x1; x1 = rotl32(x1, r); x1 ^= x0;
}

__device__ inline void threefry2x32(uint32_t k0, uint32_t k1,
                                    uint32_t c0, uint32_t c1,
                                    uint32_t& y0, uint32_t& y1) {
  const uint32_t ks2 = k0 ^ k1 ^ 0x1BD11BDAu;
  uint32_t x0 = c0 + k0;
  uint32_t x1 = c1 + k1;
  tf_r(x0,x1,13); tf_r(x0,x1,15); tf_r(x0,x1,26); tf_r(x0,x1,6);
  x0 += k1;  x1 += ks2 + 1u;
  tf_r(x0,x1,17); tf_r(x0,x1,29); tf_r(x0,x1,16); tf_r(x0,x1,24);
  x0 += ks2; x1 += k0 + 2u;
  tf_r(x0,x1,13); tf_r(x0,x1,15); tf_r(x0,x1,26); tf_r(x0,x1,6);
  x0 += k0;  x1 += k1 + 3u;
  tf_r(x0,x1,17); tf_r(x0,x1,29); tf_r(x0,x1,16); tf_r(x0,x1,24);
  x0 += k1;  x1 += ks2 + 4u;
  tf_r(x0,x1,13); tf_r(x0,x1,15); tf_r(x0,x1,26); tf_r(x0,x1,6);
  x0 += ks2; x1 += k0 + 5u;
  y0 = x0; y1 = x1;
}

// i-th element of jax random_bits(key, shape=(n,)) (n even), halves-concat order.
__device__ inline uint32_t random_bits(uint32_t k0, uint32_t k1, int i, int n) {
  const int half = n >> 1;
  uint32_t y0, y1;
  if (i < half) {
    threefry2x32(k0, k1, (uint32_t)i, (uint32_t)(i + half), y0, y1);
    return y0;
  }
  threefry2x32(k0, k1, (uint32_t)(i - half), (uint32_t)i, y0, y1);
  return y1;
}

__device__ inline float uniform01(uint32_t k0, uint32_t k1, int i, int n) {
  const uint32_t bits = random_bits(k0, k1, i, n);
  return __uint_as_float((bits >> 9) | 0x3f800000u) - 1.0f;
}

// ---------------- per-batch mask parameters ----------------
// params[b] = { t0, t1, f0, f1 } (half-open masked intervals)
__global__ void specaug_mask_params(const int* __restrict__ lengths,
                                    int4* __restrict__ params) {
  const int b = threadIdx.x;
  if (b >= kB) return;

  // root key for seed 42
  const uint32_t rk0 = 0u, rk1 = 42u;
  // kt, kf = split(root, 2): counts iota(4)
  const uint32_t kt0 = random_bits(rk0, rk1, 0, 4);
  const uint32_t kt1 = random_bits(rk0, rk1, 1, 4);
  const uint32_t kf0 = random_bits(rk0, rk1, 2, 4);
  const uint32_t kf1 = random_bits(rk0, rk1, 3, 4);

  // ---- time mask: _, k1, k2 = split(kt, 3) : counts iota(6)
  const uint32_t a10 = random_bits(kt0, kt1, 2, 6);
  const uint32_t a11 = random_bits(kt0, kt1, 3, 6);
  const uint32_t a20 = random_bits(kt0, kt1, 4, 6);
  const uint32_t a21 = random_bits(kt0, kt1, 5, 6);
  const float u1 = uniform01(a10, a11, b, kB);
  const float u2 = uniform01(a20, a21, b, kB);

  const int   L     = lengths[b];
  const int   max_t = max(1, (int)((float)L * 0.2f));
  const int   tlen  = (int)floorf(u1 * (float)max_t) + 1;
  const int   t0max = max(L - tlen, 0);
  const int   t0    = (int)floorf(u2 * (float)(t0max + 1));

  // ---- freq mask: _, k1, k2 = split(kf, 3)
  const uint32_t c10 = random_bits(kf0, kf1, 2, 6);
  const uint32_t c11 = random_bits(kf0, kf1, 3, 6);
  const uint32_t c20 = random_bits(kf0, kf1, 4, 6);
  const uint32_t c21 = random_bits(kf0, kf1, 5, 6);

  const int max_f = (int)(kA * 0.15f);                  // 192
  const uint32_t fb = random_bits(c10, c11, b, kB);
  const int   flen  = 1 + (int)(fb % (uint32_t)max_f);  // randint(1, max_f+1)
  const float u3    = uniform01(c20, c21, b, kB);
  const int   f0max = kA - flen;
  const int   f0    = (int)floorf(u3 * (float)(f0max + 1));

  params[b] = make_int4(t0, t0 + tlen, f0, f0 + flen);
}

// ---------------- streaming apply kernel ----------------
constexpr int ROWS_PB = 4;                 // rows (b,t) per block
constexpr int NTHREADS = 256;              // 8 wave32
constexpr int QROW = kD / 4;               // 384 float4 per row
constexpr int QPB = ROWS_PB * QROW;        // 1536 float4 = 24 KB per block
constexpr int ITERS = QPB / NTHREADS;      // 6 b128 per thread

#define HAS_ASYNC_LDS __has_builtin(__builtin_amdgcn_global_load_async_to_lds_b128)

__global__ __launch_bounds__(NTHREADS) void specaug_apply(
    const float* __restrict__ X, const int4* __restrict__ params,
    float* __restrict__ out) {
  __shared__ v4f tile[QPB];   // 24 KB of the 320 KB WGP LDS

  const int tid = threadIdx.x;
  const long long qbase = (long long)blockIdx.x * QPB;
  const v4f* __restrict__ src = reinterpret_cast<const v4f*>(X) + qbase;
  v4f* __restrict__ dst = reinterpret_cast<v4f*>(out) + qbase;

  const int b = blockIdx.x / (kT / ROWS_PB);           // 256 blocks per batch
  const int trow0 = (blockIdx.x % (kT / ROWS_PB)) * ROWS_PB;
  const int4 p = params[b];                            // {t0,t1,f0,f1}

#if HAS_ASYNC_LDS
  // Stage the 24 KB tile into LDS with CDNA5 async b128 copies (ASYNCcnt).
  #pragma unroll
  for (int k = 0; k < ITERS; ++k) {
    const int q = tid + k * NTHREADS;
    __attribute__((address_space(1))) v4i* gsrc =
        (__attribute__((address_space(1))) v4i*)(uintptr_t)(src + q);
    __attribute__((address_space(3))) v4i* ldst =
        (__attribute__((address_space(3))) v4i*)(uint32_t)(uintptr_t)(&tile[q]);
    __builtin_amdgcn_global_load_async_to_lds_b128(gsrc, ldst, 0, 0);
  }
  // Each lane consumes only the LDS slots it requested -> per-wave wait, no barrier.
  #if __has_builtin(__builtin_amdgcn_s_wait_asynccnt)
  __builtin_amdgcn_s_wait_asynccnt(0);
  #else
  asm volatile("s_wait_asynccnt 0" ::: "memory");
  #endif
#endif

  #pragma unroll
  for (int k = 0; k < ITERS; ++k) {
    const int q = tid + k * NTHREADS;
#if HAS_ASYNC_LDS
    v4f v = tile[q];                                   // ds_load_b128
#else
    v4f v = __builtin_nontemporal_load(src + q);       // global_load_b128 th:NT
#endif
    const int r  = q / QROW;
    const int qr = q - r * QROW;
    const int t  = trow0 + r;
    const float ts = (t >= p.x && t < p.y) ? 0.0f : 1.0f;
    if (qr >= kV / 4) {                                // audio features only
      const int a = qr * 4 - kV;
      v.x *= ts * ((a + 0 >= p.z && a + 0 < p.w) ? 0.0f : 1.0f);
      v.y *= ts * ((a + 1 >= p.z && a + 1 < p.w) ? 0.0f : 1.0f);
      v.z *= ts * ((a + 2 >= p.z && a + 2 < p.w) ? 0.0f : 1.0f);
      v.w *= ts * ((a + 3 >= p.z && a + 3 < p.w) ? 0.0f : 1.0f);
    }
    __builtin_nontemporal_store(v, dst + q);           // streaming store, keep L2 clean
  }
}

// ---------------- launch ----------------
extern "C" void kernel_launch(void* const* d_in, const int* in_sizes, int n_in,
                              void* d_out, int out_size, void* d_ws, size_t ws_size,
                              hipStream_t stream) {
  (void)in_sizes; (void)n_in; (void)out_size; (void)ws_size;
  const float* X = (const float*)d_in[0];
  const int* lengths = (const int*)d_in[1];
  float* out = (float*)d_out;
  int4* params = (int4*)d_ws;   // 32 * 16 B of scratch

  hipLaunchKernelGGL(specaug_mask_params, dim3(1), dim3(kB), 0, stream,
                     lengths, params);
  hipLaunchKernelGGL(specaug_apply, dim3(kB * kT / ROWS_PB), dim3(NTHREADS), 0,
                     stream, X, params, out);
}